// ShapleyGNNLayer_44770739093928
// MI455X (gfx1250) — compile-verified
//
#include <hip/hip_runtime.h>

typedef __attribute__((ext_vector_type(2))) float v2f;
typedef __attribute__((ext_vector_type(4))) float v4f;
typedef __attribute__((ext_vector_type(8))) float v8f;

#define F 128                 // F_IN == F_OUT == 128
#define DEG 5
#define NODES_PER_BLOCK 128
#define THREADS 256
#define KCHUNKS (F / 4)       // 32 chunks of K=4 for wmma_f32_16x16x4_f32
#define OTILES (F / 16)       // 8 output tiles of 16 channels
#define LDSTRIDE 132          // 128 + 4 pad: row stride mod 64 banks = 4 -> conflict-free b64

// Exact closed-form Shapley coefficients for d=5 (see analysis):
//   shapley = BETA * x[n] + ALPHA * sum_j x[nbr_j]
#define ALPHA (-0.048333333333333334f)   // -87/1800
#define BETA  ( 1.4083333333333334f)     // 1 + 49/120

__launch_bounds__(THREADS)
__global__ void shapley_gnn_fused(const float* __restrict__ x,
                                  const int*   __restrict__ col,   // [n*DEG] neighbor ids
                                  const float* __restrict__ W,     // [F][F] row-major: W[o][f]
                                  float*       __restrict__ out,   // [n][F]
                                  int n_nodes)
{
    __shared__ float agg [NODES_PER_BLOCK * LDSTRIDE];   // 67.6 KB, padded
    __shared__ float wlds[F * LDSTRIDE];                 // 67.6 KB, padded copy of W

    const int lane      = threadIdx.x & 31;
    const int wave      = threadIdx.x >> 5;      // 0..7
    const int l15       = lane & 15;
    const int half      = lane >> 4;             // 0 or 1
    const int node_base = blockIdx.x * NODES_PER_BLOCK;
    const int wrow0     = wave * 16;             // this wave's local row base in agg

    // ---------------- Stage W into LDS (block-cooperative, coalesced) --------
    {
        const int o  = threadIdx.x >> 1;               // 0..127: output channel row
        const int cb = (threadIdx.x & 1) * 64;         // half-row column base
        const float* src = W + (size_t)o * F + cb;
        float*       dst = wlds + o * LDSTRIDE + cb;
        #pragma unroll
        for (int c = 0; c < 64; c += 4)
            *(v4f*)(dst + c) = *(const v4f*)(src + c); // global b128 -> ds b128
    }

    // ---------------- Phase 1: gather + combine into LDS ---------------------
    // One wave loads one full 512B row per instruction (32 lanes x float4).
    #pragma unroll 4
    for (int r = 0; r < 16; ++r) {
        const int lrow   = wrow0 + r;
        int node         = node_base + lrow;
        const int nclamp = node < n_nodes ? node : (n_nodes - 1);

        const float* xs = x + (size_t)nclamp * F + lane * 4;
        v4f a = *(const v4f*)xs;                 // coalesced: block's own rows are contiguous
        a.x *= BETA; a.y *= BETA; a.z *= BETA; a.w *= BETA;

        #pragma unroll
        for (int j = 0; j < DEG; ++j) {
            const int nb = col[nclamp * DEG + j];         // uniform across wave -> 1 request
            const float* xn = x + (size_t)nb * F + lane * 4;
            v4f u = *(const v4f*)xn;                      // random row, but 512B contiguous
            a.x += ALPHA * u.x; a.y += ALPHA * u.y;
            a.z += ALPHA * u.z; a.w += ALPHA * u.w;
        }
        *(v4f*)&agg[lrow * LDSTRIDE + lane * 4] = a;
    }

    __syncthreads();   // wlds visible to all waves (agg rows are wave-private anyway)

    // ---------------- Phase 2: out = relu(agg @ W^T) via f32 WMMA -------------
    // A tile (16x4):  lane (l15) = row M, VGPR0/1 = K = kc*4 + 2*half (+1)
    // B tile (4x16):  lane (l15) = col N (= output channel), same K addressing,
    //                 value = W[o = t*16 + l15][k] -> consecutive pair = ds b64.
    // Padded stride => each b64 hits 32 distinct banks across the wave.
    const float* aggrow = &agg [(wrow0 + l15) * LDSTRIDE + 2 * half];
    const float* wrow   = &wlds[ l15          * LDSTRIDE + 2 * half];

    v8f acc[OTILES] = {};
    #pragma unroll 4
    for (int kc = 0; kc < KCHUNKS; ++kc) {
        const int kcol = kc * 4;
        v2f a = *(const v2f*)(aggrow + kcol);             // ds_load_b64, conflict-free
        #pragma unroll
        for (int t = 0; t < OTILES; ++t) {
            v2f b = *(const v2f*)(wrow + t * 16 * LDSTRIDE + kcol);   // ds_load_b64
            acc[t] = __builtin_amdgcn_wmma_f32_16x16x4_f32(
                /*neg_a=*/false, a, /*neg_b=*/false, b,
                /*c_mod=*/(short)0, acc[t],
                /*reuse_a=*/false, /*reuse_b=*/false);
        }
    }

    // ---------------- Epilogue: ReLU + predicated stores ----------------------
    // D layout: VGPR r -> M = r + 8*half, N = l15 (within tile t).
    #pragma unroll
    for (int t = 0; t < OTILES; ++t) {
        #pragma unroll
        for (int r = 0; r < 8; ++r) {
            const int m    = r + 8 * half;
            const int node = node_base + wrow0 + m;
            if (node < n_nodes) {
                float v = acc[t][r];
                out[(size_t)node * F + t * 16 + l15] = v > 0.0f ? v : 0.0f;
            }
        }
    }
}

extern "C" void kernel_launch(void* const* d_in, const int* in_sizes, int n_in,
                              void* d_out, int out_size, void* d_ws, size_t ws_size,
                              hipStream_t stream) {
    const float* x          = (const float*)d_in[0];           // [n, 128] f32
    const int*   edge_index = (const int*)  d_in[1];           // [2, n*5] i32
    const float* W          = (const float*)d_in[2];           // [128, 128] f32
    float*       out        = (float*)d_out;                   // [n, 128] f32

    const int n   = in_sizes[0] / F;
    const int*col = edge_index + in_sizes[1] / 2;              // second row of edge_index

    const int blocks = (n + NODES_PER_BLOCK - 1) / NODES_PER_BLOCK;
    shapley_gnn_fused<<<blocks, THREADS, 0, stream>>>(x, col, W, out, n);
}